// WassersteinLoss_73057393705633
// MI455X (gfx1250) — compile-verified
//
#include <hip/hip_runtime.h>
#include <hip/hip_bf16.h>

#define BATCH 8
#define NPTS  2048
#define LDIAG (NPTS - 1)
#define BIGF  1e30f
#define EPSF  1e-12f

typedef __attribute__((ext_vector_type(2))) float v2f;
typedef __attribute__((ext_vector_type(8))) float v8f;

// ---------------------------------------------------------------------------
// squared norms per point: sq[i] = |P_i|^2 over all BATCH*NPTS points
// ---------------------------------------------------------------------------
__global__ void sqnorm_kernel(const float* __restrict__ P, float* __restrict__ sq) {
    int i = blockIdx.x * blockDim.x + threadIdx.x;
    if (i < BATCH * NPTS) {
        float x = P[3 * i + 0], y = P[3 * i + 1], z = P[3 * i + 2];
        sq[i] = x * x + y * y + z * z;
    }
}

// ---------------------------------------------------------------------------
// Distance matrix via WMMA Gram tiles:
//   D[b][i][j] = sqrt(max(sq[i] + sq[j] - 2 * (P_i . P_j), EPS))
// One wave computes one 16x16 tile with V_WMMA_F32_16X16X4_F32 (K padded 3->4).
// A-frag layout (16x4 f32): lane l: M = l%16, K = 2*(l/16) + vgpr.
// C/D layout: vgpr r, lane l -> (M = r + 8*(l/16), N = l%16).
// ---------------------------------------------------------------------------
__global__ void dist_kernel(const float* __restrict__ P,
                            const float* __restrict__ sq,
                            float* __restrict__ D) {
    const int b    = blockIdx.y;
    const int lane = threadIdx.x & 31;
    const int wv   = threadIdx.x >> 5;
    const int tile = blockIdx.x * (blockDim.x >> 5) + wv;   // 0 .. 16383
    const int ti   = tile >> 7;                             // 128 tiles per dim
    const int tj   = tile & 127;
    const int I    = ti * 16;
    const int J    = tj * 16;
    const int m    = lane & 15;
    const int g    = lane >> 4;                             // K half select

    const float* Pb  = P  + (size_t)b * NPTS * 3;
    const float* sqb = sq + (size_t)b * NPTS;

    v2f a, bb;
    {
        int row = I + m;
        a.x = Pb[row * 3 + 2 * g];                      // K = 0 or 2
        a.y = (g == 0) ? Pb[row * 3 + 1] : 0.0f;        // K = 1, or 3 (zero pad)
    }
    {
        int row = J + m;
        bb.x = Pb[row * 3 + 2 * g];
        bb.y = (g == 0) ? Pb[row * 3 + 1] : 0.0f;
    }

    v8f c = {0.f, 0.f, 0.f, 0.f, 0.f, 0.f, 0.f, 0.f};
    // (neg_a, A, neg_b, B, c_mod, C, reuse_a, reuse_b)
    c = __builtin_amdgcn_wmma_f32_16x16x4_f32(false, a, false, bb,
                                              (short)0, c, false, false);

    float* Db = D + (size_t)b * NPTS * NPTS;
    const int   gj = J + m;
    const float sj = sqb[gj];
#pragma unroll
    for (int r = 0; r < 8; ++r) {
        int   gi = I + r + 8 * g;
        float v  = sqb[gi] + sj - 2.0f * c[r];
        Db[(size_t)gi * NPTS + gj] = sqrtf(fmaxf(v, EPSF));
    }
}

// ---------------------------------------------------------------------------
// Prim's MST per batch: one block (1024 threads = 32 waves) per batch.
// mind/visited live in LDS; row D[j] streams from the (L2-resident) matrix.
// Argmin keeps lowest index on ties (JAX argmin semantics).
// ---------------------------------------------------------------------------
__global__ void __launch_bounds__(1024)
prim_kernel(const float* __restrict__ D, float* __restrict__ deaths) {
    const int b   = blockIdx.x;
    const int tid = threadIdx.x;
    const int lane = tid & 31, wv = tid >> 5;

    __shared__ float         mind[NPTS];
    __shared__ unsigned char vis[NPTS];
    __shared__ float rv[32];
    __shared__ int   ri[32];
    __shared__ int   sj;

    const float* Db = D + (size_t)b * NPTS * NPTS;

    for (int k = tid; k < NPTS; k += blockDim.x) {
        mind[k] = Db[k];
        vis[k]  = (k == 0) ? 1 : 0;
    }
    __syncthreads();

    for (int s = 0; s < NPTS - 1; ++s) {
        float best = BIGF;
        int   bi   = 0x7fffffff;
        for (int k = tid; k < NPTS; k += blockDim.x) {
            float v = vis[k] ? BIGF : mind[k];
            if (v < best) { best = v; bi = k; }
        }
        for (int off = 16; off > 0; off >>= 1) {
            float ov = __shfl_xor(best, off, 32);
            int   oi = __shfl_xor(bi,   off, 32);
            if (ov < best || (ov == best && oi < bi)) { best = ov; bi = oi; }
        }
        if (lane == 0) { rv[wv] = best; ri[wv] = bi; }
        __syncthreads();
        if (wv == 0) {
            best = rv[lane]; bi = ri[lane];
            for (int off = 16; off > 0; off >>= 1) {
                float ov = __shfl_xor(best, off, 32);
                int   oi = __shfl_xor(bi,   off, 32);
                if (ov < best || (ov == best && oi < bi)) { best = ov; bi = oi; }
            }
            if (lane == 0) {
                sj = bi;
                vis[bi] = 1;
                deaths[(size_t)b * LDIAG + s] = best;
            }
        }
        __syncthreads();
        const float* row = Db + (size_t)sj * NPTS;
        for (int k = tid; k < NPTS; k += blockDim.x)
            mind[k] = fminf(mind[k], row[k]);
        __syncthreads();
    }
}

// ---------------------------------------------------------------------------
// Greedy matching + closed-form norm, one block per batch.
// cost[i][j] = (dX_i - dY_j)^2 ; row i takes cheapest available column.
// sum_i (dX_i - m)^2 = S2 - 2 m S1 + L m^2  (no O(L^2) matrix needed).
// ---------------------------------------------------------------------------
__global__ void __launch_bounds__(1024)
match_kernel(const float* __restrict__ dX, const float* __restrict__ dY,
             float* __restrict__ psum) {
    const int b   = blockIdx.x;
    const int tid = threadIdx.x;
    const int lane = tid & 31, wv = tid >> 5;

    __shared__ float         lX[LDIAG];
    __shared__ float         lY[LDIAG];
    __shared__ unsigned char avail[LDIAG];
    __shared__ float rv[32], r1[32], r2[32];
    __shared__ int   ri[32];
    __shared__ int   sj;
    __shared__ float sS1, sS2;

    float s1 = 0.f, s2 = 0.f;
    for (int k = tid; k < LDIAG; k += blockDim.x) {
        float x = dX[(size_t)b * LDIAG + k];
        lX[k] = x;
        lY[k] = dY[(size_t)b * LDIAG + k];
        avail[k] = 1;
        s1 += x;
        s2 += x * x;
    }
    for (int off = 16; off > 0; off >>= 1) {
        s1 += __shfl_xor(s1, off, 32);
        s2 += __shfl_xor(s2, off, 32);
    }
    if (lane == 0) { r1[wv] = s1; r2[wv] = s2; }
    __syncthreads();
    if (wv == 0) {
        s1 = r1[lane]; s2 = r2[lane];
        for (int off = 16; off > 0; off >>= 1) {
            s1 += __shfl_xor(s1, off, 32);
            s2 += __shfl_xor(s2, off, 32);
        }
        if (lane == 0) { sS1 = s1; sS2 = s2; }
    }
    __syncthreads();
    const float S1 = sS1, S2 = sS2;

    float acc = 0.f;
    for (int i = 0; i < LDIAG; ++i) {
        const float xi = lX[i];
        float best = BIGF;
        int   bi   = 0x7fffffff;
        for (int k = tid; k < LDIAG; k += blockDim.x) {
            float dv = xi - lY[k];
            float v  = avail[k] ? dv * dv : BIGF;
            if (v < best) { best = v; bi = k; }
        }
        for (int off = 16; off > 0; off >>= 1) {
            float ov = __shfl_xor(best, off, 32);
            int   oi = __shfl_xor(bi,   off, 32);
            if (ov < best || (ov == best && oi < bi)) { best = ov; bi = oi; }
        }
        if (lane == 0) { rv[wv] = best; ri[wv] = bi; }
        __syncthreads();
        if (wv == 0) {
            best = rv[lane]; bi = ri[lane];
            for (int off = 16; off > 0; off >>= 1) {
                float ov = __shfl_xor(best, off, 32);
                int   oi = __shfl_xor(bi,   off, 32);
                if (ov < best || (ov == best && oi < bi)) { best = ov; bi = oi; }
            }
            if (lane == 0) { sj = bi; avail[bi] = 0; }
        }
        __syncthreads();
        if (tid == 0) {
            float m = lY[sj];
            float t = S2 - 2.0f * m * S1 + (float)LDIAG * m * m;
            acc += sqrtf(fmaxf(t, 0.0f));
        }
    }
    if (tid == 0) psum[b] = acc;
}

// Deterministic final sum over the 8 per-batch partials.
__global__ void finalize_kernel(const float* __restrict__ psum, float* __restrict__ out) {
    float s = 0.f;
#pragma unroll
    for (int b = 0; b < BATCH; ++b) s += psum[b];
    out[0] = s;
}

// ---------------------------------------------------------------------------
extern "C" void kernel_launch(void* const* d_in, const int* in_sizes, int n_in,
                              void* d_out, int out_size, void* d_ws, size_t ws_size,
                              hipStream_t stream) {
    const float* X = (const float*)d_in[0];   // [8, 2048, 3]
    const float* Y = (const float*)d_in[1];   // [8, 2048, 3]
    float* out = (float*)d_out;               // scalar

    // workspace layout (bytes)
    const size_t D_BYTES   = (size_t)BATCH * NPTS * NPTS * sizeof(float); // 134 MB
    const size_t SQ_BYTES  = (size_t)BATCH * NPTS * sizeof(float);
    const size_t DTH_BYTES = (size_t)BATCH * LDIAG * sizeof(float);

    char*  ws      = (char*)d_ws;
    float* Dws     = (float*)(ws);
    float* sqws    = (float*)(ws + D_BYTES);
    float* deathsX = (float*)(ws + D_BYTES + SQ_BYTES);
    float* deathsY = (float*)(ws + D_BYTES + SQ_BYTES + DTH_BYTES);
    float* psum    = (float*)(ws + D_BYTES + SQ_BYTES + 2 * DTH_BYTES);

    const int sq_blocks = (BATCH * NPTS + 255) / 256;
    // 16384 tiles per batch, 8 waves (256 threads) per block -> 2048 blocks/batch
    dim3 dist_grid(2048, BATCH);

    // ---- X pipeline ----
    sqnorm_kernel<<<sq_blocks, 256, 0, stream>>>(X, sqws);
    dist_kernel<<<dist_grid, 256, 0, stream>>>(X, sqws, Dws);
    prim_kernel<<<BATCH, 1024, 0, stream>>>(Dws, deathsX);

    // ---- Y pipeline (reuses D and sq buffers; stream ordering guarantees safety) ----
    sqnorm_kernel<<<sq_blocks, 256, 0, stream>>>(Y, sqws);
    dist_kernel<<<dist_grid, 256, 0, stream>>>(Y, sqws, Dws);
    prim_kernel<<<BATCH, 1024, 0, stream>>>(Dws, deathsY);

    // ---- matching + reduction ----
    match_kernel<<<BATCH, 1024, 0, stream>>>(deathsX, deathsY, psum);
    finalize_kernel<<<1, 1, 0, stream>>>(psum, out);
}